// RNN_simple_33603824123978
// MI455X (gfx1250) — compile-verified
//
#include <hip/hip_runtime.h>
#include <stdint.h>

#define T_STEPS 30
#define HID     20
#define BLK     256                 // threads per block == batch elements per block
#define TILE_F  (BLK * T_STEPS)     // floats per x/y tile = 7680 (30 KB)

typedef __attribute__((ext_vector_type(4))) unsigned int v4u;
typedef __attribute__((ext_vector_type(8))) int          v8i;
typedef __attribute__((ext_vector_type(4))) int          v4i;

#if defined(__has_builtin)
#if __has_builtin(__builtin_amdgcn_tensor_load_to_lds) && __has_builtin(__builtin_amdgcn_tensor_store_from_lds)
#define HAVE_TDM 1
#endif
#if __has_builtin(__builtin_amdgcn_tanhf)
#define HAVE_TANH 1
#endif
#endif
#ifndef HAVE_TDM
#define HAVE_TDM 0
#endif
#ifndef HAVE_TANH
#define HAVE_TANH 0
#endif

__device__ __forceinline__ float fast_tanh(float v) {
#if HAVE_TANH
  return __builtin_amdgcn_tanhf(v);   // v_tanh_f32 (CDNA5 TRANS op)
#else
  return tanhf(v);
#endif
}

#if HAVE_TDM
// Build a D# for a 1-D contiguous f32 copy of n elements (n < 65536 so it fits
// tile_dim0[15:0]).  Field packing per CDNA5 ISA:
//   group0: dw0 = count=1, dw1 = lds byte addr, dw2 = gaddr[31:0],
//           dw3 = gaddr[56:32] | type=2 << 30
//   group1: dw0 = data_size(4B)=2 << 16
//           dw1 = tensor_dim0[15:0]  << 16          (bits 63:48)
//           dw2 = tensor_dim0[31:16] | tensor_dim1(=1) << 16
//           dw3 = tensor_dim1[31:16](=0) | tile_dim0 << 16
//           dw4 = tile_dim1(=1) | tile_dim2(=0) << 16
//           dw5 = tensor_dim0_stride[31:0] = n ; dw6 = dw7 = 0
__device__ __forceinline__ void tdm_desc(uint32_t lds_off, const void* gptr,
                                         uint32_t n, v4u& g0, v8i& g1) {
  uint64_t ga = (uint64_t)(uintptr_t)gptr;
  g0.x = 1u;                                        // count = 1 (valid)
  g0.y = lds_off;                                   // LDS byte address
  g0.z = (uint32_t)ga;                              // global addr lo
  g0.w = ((uint32_t)(ga >> 32) & 0x1FFFFFFu) | (2u << 30);  // addr hi | type=2
  g1[0] = (int)(2u << 16);                          // data_size = 4 bytes
  g1[1] = (int)((n & 0xFFFFu) << 16);               // tensor_dim0 lo16
  g1[2] = (int)((n >> 16) | (1u << 16));            // tensor_dim0 hi16 | tensor_dim1=1
  g1[3] = (int)((n & 0xFFFFu) << 16);               // tile_dim0 = n
  g1[4] = 1;                                        // tile_dim1 = 1, tile_dim2 = 0
  g1[5] = (int)n;                                   // tensor_dim0_stride = n
  g1[6] = 0;
  g1[7] = 0;
}

__device__ __forceinline__ void tdm_load_1d(uint32_t lds_off, const void* gptr, uint32_t n) {
  v4u g0; v8i g1; tdm_desc(lds_off, gptr, n, g0, g1);
  v4i z4 = {0, 0, 0, 0};
#if __clang_major__ >= 23
  v8i z8 = {0, 0, 0, 0, 0, 0, 0, 0};
  __builtin_amdgcn_tensor_load_to_lds(g0, g1, z4, z4, z8, 0);
#else
  __builtin_amdgcn_tensor_load_to_lds(g0, g1, z4, z4, 0);
#endif
}

__device__ __forceinline__ void tdm_store_1d(uint32_t lds_off, void* gptr, uint32_t n) {
  v4u g0; v8i g1; tdm_desc(lds_off, gptr, n, g0, g1);
  v4i z4 = {0, 0, 0, 0};
#if __clang_major__ >= 23
  v8i z8 = {0, 0, 0, 0, 0, 0, 0, 0};
  __builtin_amdgcn_tensor_store_from_lds(g0, g1, z4, z4, z8, 0);
#else
  __builtin_amdgcn_tensor_store_from_lds(g0, g1, z4, z4, 0);
#endif
}
#endif  // HAVE_TDM

__global__ void __launch_bounds__(BLK)
rnn_mlp_kernel(const float* __restrict__ x,  const float* __restrict__ W1,
               const float* __restrict__ b1, const float* __restrict__ W2,
               const float* __restrict__ b2, float* __restrict__ y, int total) {
  __shared__ __align__(16) float sx[TILE_F];
  __shared__ __align__(16) float sy[TILE_F];

  const int  tid  = threadIdx.x;
  const long base = (long)blockIdx.x * TILE_F;          // float index of this tile
  const int  n    = (int)min((long)TILE_F, (long)total - base);  // multiple of 30

  // ---- Stage x tile into LDS via the Tensor Data Mover (wave 0 issues) ----
#if HAVE_TDM
  if (tid == 0) {
    tdm_load_1d((uint32_t)(uintptr_t)&sx[0], x + base, (uint32_t)n);
    __builtin_amdgcn_s_wait_tensorcnt(0);
  }
#else
  for (int i = tid; i < n; i += BLK) sx[i] = x[base + i];
#endif
  __syncthreads();

  const bool active = (tid * T_STEPS) < n;

  // Pull this lane's 30 consecutive x values (8-byte aligned float2 LDS loads).
  float xv[T_STEPS];
  if (active) {
#pragma unroll
    for (int i = 0; i < T_STEPS / 2; ++i) {
      float2 v = *(const float2*)&sx[tid * T_STEPS + 2 * i];
      xv[2 * i] = v.x; xv[2 * i + 1] = v.y;
    }
  }

  float yv[T_STEPS];
  if (active) {
    // Keep this loop rolled: one iteration's 61 parameter floats fit in SGPRs
    // (scalar/SMEM path); unrolling x2 overflows the 106-SGPR file and demotes
    // the parameter loads to per-lane VMEM.
    for (int t = 0; t < T_STEPS; ++t) {
      // t is wave-uniform -> these become scalar (SMEM) loads into SGPRs.
      const float* w1t = W1 + t * HID;
      const float* b1t = b1 + t * HID;
      const float* w2t = W2 + t * HID;
      const float  bt2 = b2[t];
      const float  xt  = xv[t];
      float a0 = 0.f, a1 = 0.f, a2 = 0.f, a3 = 0.f;
#pragma unroll
      for (int h = 0; h < HID; h += 4) {
        float h0 = fmaxf(fmaf(xt, w1t[h + 0], b1t[h + 0]), 0.f);
        float h1 = fmaxf(fmaf(xt, w1t[h + 1], b1t[h + 1]), 0.f);
        float h2 = fmaxf(fmaf(xt, w1t[h + 2], b1t[h + 2]), 0.f);
        float h3 = fmaxf(fmaf(xt, w1t[h + 3], b1t[h + 3]), 0.f);
        a0 = fmaf(h0, w2t[h + 0], a0);
        a1 = fmaf(h1, w2t[h + 1], a1);
        a2 = fmaf(h2, w2t[h + 2], a2);
        a3 = fmaf(h3, w2t[h + 3], a3);
      }
      yv[t] = fast_tanh((a0 + a1) + (a2 + a3) + bt2);
    }
    // Stage results into LDS for the bulk DMA store.
#pragma unroll
    for (int i = 0; i < T_STEPS / 2; ++i) {
      *(float2*)&sy[tid * T_STEPS + 2 * i] = make_float2(yv[2 * i], yv[2 * i + 1]);
    }
  }
  __syncthreads();

  // ---- Stream y tile back out via TDM ----
#if HAVE_TDM
  if (tid == 0) {
    tdm_store_1d((uint32_t)(uintptr_t)&sy[0], y + base, (uint32_t)n);
    __builtin_amdgcn_s_wait_tensorcnt(0);
  }
#else
  for (int i = tid; i < n; i += BLK) y[base + i] = sy[i];
#endif
}

extern "C" void kernel_launch(void* const* d_in, const int* in_sizes, int n_in,
                              void* d_out, int out_size, void* d_ws, size_t ws_size,
                              hipStream_t stream) {
  const float* x  = (const float*)d_in[0];   // [B,T,1]
  const float* W1 = (const float*)d_in[1];   // [T,H,1]
  const float* b1 = (const float*)d_in[2];   // [T,H]
  const float* W2 = (const float*)d_in[3];   // [T,1,H]
  const float* b2 = (const float*)d_in[4];   // [T,1]
  float*       y  = (float*)d_out;           // [B,T,1]

  const int total   = in_sizes[0];           // B*T
  const int nblocks = (total + TILE_F - 1) / TILE_F;
  rnn_mlp_kernel<<<nblocks, BLK, 0, stream>>>(x, W1, b1, W2, b2, y, total);
}